// MoEsMultiTaskModel_86174223827165
// MI455X (gfx1250) — compile-verified
//
#include <hip/hip_runtime.h>

// MoE multi-task model, fully fused for MI455X (gfx1250, wave32).
// GEMMs run on V_WMMA_F32_16X16X4_F32 (fp32 matrix pipe, matches fp32 reference).
//
// Shapes: B=16384, S=50, EMBED=128, HID=256, E=16 experts.
// One block = 128 threads (4 waves) = one 16-row batch tile.

typedef float v2f __attribute__((ext_vector_type(2)));
typedef float v8f __attribute__((ext_vector_type(8)));

#define EMBED 128
#define HID   256
#define NEXP  16
#define SEQ   50
#define TB    16          // batch rows per block
#define HPAD  132         // EMBED + 4 dwords (bank-conflict padding: 132 % 64 == 4)
#define EPAD  260         // HID   + 4 dwords (260 % 64 == 4)

__global__ __launch_bounds__(128)
void moe_fused_kernel(const int*   __restrict__ x,
                      const float* __restrict__ emb,
                      const float* __restrict__ gw,
                      const float* __restrict__ gb,
                      const float* __restrict__ w1,
                      const float* __restrict__ b1,
                      const float* __restrict__ w2,
                      const float* __restrict__ b2,
                      const float* __restrict__ tw,
                      const float* __restrict__ tb,
                      float*       __restrict__ out,
                      int NC)
{
    __shared__ float hT[TB * HPAD];       // mean embedding tile [16][128] (padded)
    __shared__ float e1T[TB * EPAD];      // expert hidden tile  [16][256] (padded), reused for combined
    __shared__ float gateT[TB * NEXP];    // gate logits -> softmax probs
    __shared__ int   idxT[TB * SEQ];      // token indices for this tile

    const int t    = threadIdx.x;
    const int lane = t & 31;
    const int half = (lane >> 4) & 1;     // which half-wave (K sub-slice for wmma frags)
    const int c15  = lane & 15;           // column within 16-wide tile
    const int wave = t >> 5;              // 0..3
    const int b0   = blockIdx.x * TB;

    // ---- Stage 1: token indices (contiguous across the tile) ----
    for (int i = t; i < TB * SEQ; i += 128)
        idxT[i] = x[(size_t)b0 * SEQ + i];
    __syncthreads();

    // ---- Stage 2: h = mean_s emb[x]; thread t owns embed dim d = t ----
    {
        const int d = t;
        for (int m = 0; m < TB; ++m) {
            float s0 = 0.f, s1 = 0.f;
            #pragma unroll 5
            for (int s = 0; s < SEQ; s += 2) {
                s0 += emb[(size_t)idxT[m * SEQ + s    ] * EMBED + d];
                s1 += emb[(size_t)idxT[m * SEQ + s + 1] * EMBED + d];
            }
            hT[m * HPAD + d] = (s0 + s1) * (1.0f / SEQ);
        }
    }
    __syncthreads();

    // ---- Stage 3: gate logits (256 (m,e) pairs, 2 per thread) ----
    for (int p = t; p < TB * NEXP; p += 128) {
        const int m = p >> 4, e = p & 15;
        float s = gb[e];
        for (int d = 0; d < EMBED; ++d)
            s += hT[m * HPAD + d] * gw[d * NEXP + e];
        gateT[p] = s;
    }
    __syncthreads();

    // softmax per row, one row per thread (t < 16); no private arrays -> no scratch
    if (t < TB) {
        float mx = -3.0e38f;
        for (int e = 0; e < NEXP; ++e) mx = fmaxf(mx, gateT[t * NEXP + e]);
        float sum = 0.f;
        for (int e = 0; e < NEXP; ++e) sum += __expf(gateT[t * NEXP + e] - mx);
        const float inv = 1.0f / sum;
        for (int e = 0; e < NEXP; ++e)
            gateT[t * NEXP + e] = __expf(gateT[t * NEXP + e] - mx) * inv;
    }
    __syncthreads();

    // ---- Stage 4: expert loop; combined accumulated in registers ----
    v8f comb[2] = { {}, {} };   // this wave owns eo columns [wave*32, wave*32+32)

    for (int e = 0; e < NEXP; ++e) {
        const float* w1e = w1 + (size_t)e * EMBED * HID;   // [128][256] row-major (K x N)
        const float* w2e = w2 + (size_t)e * HID * EMBED;   // [256][128] row-major (K x N)

        // GEMM1: e1[16,256] = relu(h @ w1e + b1[e]); wave owns 4 of 16 N-tiles
        v8f acc[4] = { {}, {}, {}, {} };
        for (int k = 0; k < EMBED / 4; ++k) {
            const int kr = 4 * k + 2 * half;
            v2f a;                                   // A frag: 16x4 fp32, lanes=rows
            a.x = hT[c15 * HPAD + kr];
            a.y = hT[c15 * HPAD + kr + 1];
            #pragma unroll
            for (int j = 0; j < 4; ++j) {
                const int col = (wave * 4 + j) * 16 + c15;
                v2f bf;                              // B frag: 4x16 fp32, lanes=cols
                bf.x = w1e[(size_t)kr * HID + col];
                bf.y = w1e[(size_t)(kr + 1) * HID + col];
                acc[j] = __builtin_amdgcn_wmma_f32_16x16x4_f32(
                             false, a, false, bf, (short)0, acc[j], false, false);
            }
        }
        // bias + relu, spill to LDS for the K-side of GEMM2
        #pragma unroll
        for (int j = 0; j < 4; ++j) {
            const int col  = (wave * 4 + j) * 16 + c15;
            const float bb = b1[e * HID + col];
            #pragma unroll
            for (int r = 0; r < 8; ++r) {
                const float v = acc[j][r] + bb;
                e1T[(r + 8 * half) * EPAD + col] = fmaxf(v, 0.0f);
            }
        }
        __syncthreads();

        // GEMM2: eo[16,128] = e1 @ w2e; wave owns 2 of 8 N-tiles
        v8f acc2[2] = { {}, {} };
        for (int k = 0; k < HID / 4; ++k) {
            const int kr = 4 * k + 2 * half;
            v2f a;
            a.x = e1T[c15 * EPAD + kr];
            a.y = e1T[c15 * EPAD + kr + 1];
            #pragma unroll
            for (int j = 0; j < 2; ++j) {
                const int col = (wave * 2 + j) * 16 + c15;
                v2f bf;
                bf.x = w2e[(size_t)kr * EMBED + col];
                bf.y = w2e[(size_t)(kr + 1) * EMBED + col];
                acc2[j] = __builtin_amdgcn_wmma_f32_16x16x4_f32(
                              false, a, false, bf, (short)0, acc2[j], false, false);
            }
        }
        // gate-weighted accumulate: combined += gate[m,e] * (eo + b2[e])
        #pragma unroll
        for (int j = 0; j < 2; ++j) {
            const int col  = (wave * 2 + j) * 16 + c15;
            const float bb = b2[e * EMBED + col];
            #pragma unroll
            for (int r = 0; r < 8; ++r) {
                const float g = gateT[(r + 8 * half) * NEXP + e];
                comb[j][r] += g * (acc2[j][r] + bb);
            }
        }
        __syncthreads();   // e1T WAR protection before next expert
    }

    // ---- Stage 5: stash combined into LDS (reuse e1T), then task head ----
    #pragma unroll
    for (int j = 0; j < 2; ++j) {
        const int col = (wave * 2 + j) * 16 + c15;
        #pragma unroll
        for (int r = 0; r < 8; ++r)
            e1T[(r + 8 * half) * EPAD + col] = comb[j][r];
    }
    __syncthreads();

    for (int p = t; p < TB * NC; p += 128) {
        const int m = p / NC, c = p % NC;
        float s = tb[c];
        for (int d = 0; d < EMBED; ++d)
            s += e1T[m * EPAD + d] * tw[d * NC + c];
        out[(size_t)(b0 + m) * NC + c] = s;
    }
}

extern "C" void kernel_launch(void* const* d_in, const int* in_sizes, int n_in,
                              void* d_out, int out_size, void* d_ws, size_t ws_size,
                              hipStream_t stream) {
    // setup_inputs order:
    // 0:x 1:task_id 2:emb 3:gate_w 4:gate_b 5:w1 6:b1 7:w2 8:b2 9:t1_w 10:t1_b 11:t2_w 12:t2_b
    const int*   x   = (const int*)  d_in[0];
    const float* emb = (const float*)d_in[2];
    const float* gw  = (const float*)d_in[3];
    const float* gb  = (const float*)d_in[4];
    const float* w1  = (const float*)d_in[5];
    const float* b1  = (const float*)d_in[6];
    const float* w2  = (const float*)d_in[7];
    const float* b2  = (const float*)d_in[8];

    const int B  = in_sizes[0] / SEQ;        // 16384
    const int NC = out_size / B;             // 10 -> task 1, 5 -> task 2
    const float* tw = (NC == 10) ? (const float*)d_in[9]  : (const float*)d_in[11];
    const float* tb = (NC == 10) ? (const float*)d_in[10] : (const float*)d_in[12];

    dim3 grid(B / TB), block(128);
    moe_fused_kernel<<<grid, block, 0, stream>>>(x, emb, gw, gb, w1, b1, w2, b2,
                                                 tw, tb, (float*)d_out, NC);
}